// Stage_75419625717819
// MI455X (gfx1250) — compile-verified
//
#include <hip/hip_runtime.h>

// ---------------------------------------------------------------------------
// Problem constants (match reference)
// ---------------------------------------------------------------------------
constexpr int B_  = 4;
constexpr int N_  = 8192;
constexpr int C_  = 64;
constexpr int S_  = 2048;
constexpr int G_  = 32;
constexpr int C2_ = 128;
constexpr int EXP_ = 4;

typedef __attribute__((ext_vector_type(2))) float v2f;
typedef __attribute__((ext_vector_type(8))) float v8f;
typedef unsigned int u32x4 __attribute__((ext_vector_type(4)));
typedef int i32x4 __attribute__((ext_vector_type(4)));
typedef int i32x8 __attribute__((ext_vector_type(8)));

__device__ __forceinline__ v8f wmma4(v2f a, v2f b, v8f c) {
  // D(16x16,f32) = A(16x4,f32) * B(4x16,f32) + C
  return __builtin_amdgcn_wmma_f32_16x16x4_f32(false, a, false, b, (short)0, c,
                                               false, false);
}

// ---------------------------------------------------------------------------
// Tensor Data Mover: 1-D contiguous copy global -> LDS (nelem 4-byte words).
// D# group0: count=1 | lds_addr | global_addr(57b) | type=2.
// D# group1: data_size=4B, tensor_dim0 = tile_dim0 = nelem (<=65535),
//            tensor_dim1 = tile_dim1 = 1, strides = nelem.
// Issued by one wave; completion via s_wait_tensorcnt 0 + block barrier.
// ---------------------------------------------------------------------------
__device__ __forceinline__ void tdm_load_1d(unsigned ldsOff, const void* gptr,
                                            unsigned nelem) {
  unsigned long long ga = (unsigned long long)(uintptr_t)gptr;
  u32x4 g0;
  g0[0] = 1u;                                            // count=1 (user mode)
  g0[1] = ldsOff;                                        // lds_addr (bytes)
  g0[2] = (unsigned)(ga & 0xFFFFFFFFull);                // global_addr[31:0]
  g0[3] = (unsigned)((ga >> 32) & 0x01FFFFFFull) | (2u << 30);  // [56:32]|type=2
  i32x8 g1;
  g1[0] = (int)(2u << 16);                               // data_size = 4B
  g1[1] = (int)((nelem & 0xFFFFu) << 16);                // tensor_dim0[15:0]
  g1[2] = (int)(((nelem >> 16) & 0xFFFFu) | (1u << 16)); // td0[31:16] | td1 lo=1
  g1[3] = (int)((nelem & 0xFFFFu) << 16);                // td1 hi=0 | tile_dim0
  g1[4] = 1;                                             // tile_dim1=1, tile_dim2=0
  g1[5] = (int)nelem;                                    // stride0[31:0]
  g1[6] = (int)((nelem & 0xFFFFu) << 16);                // stride0 hi=0|stride1 lo
  g1[7] = (int)(nelem >> 16);                            // stride1[47:16]
  i32x4 z4 = {0, 0, 0, 0};
#if defined(__clang_major__) && __clang_major__ >= 23
  i32x8 z8 = {0, 0, 0, 0, 0, 0, 0, 0};
  __builtin_amdgcn_tensor_load_to_lds(g0, g1, z4, z4, z8, 0);
#else
  __builtin_amdgcn_tensor_load_to_lds(g0, g1, z4, z4, 0);
#endif
}

// ---------------------------------------------------------------------------
// 1) Layout prep: (B,3,N)->(B,N,3) coords, (B,C,N)->(B,N,C) features
// ---------------------------------------------------------------------------
__global__ __launch_bounds__(256) void prep_kernel(
    const float* __restrict__ pcoor, const float* __restrict__ pfea,
    float* __restrict__ coorW, float* __restrict__ feaW) {
  int gid = blockIdx.x * blockDim.x + threadIdx.x;   // over B*N
  if (gid >= B_ * N_) return;
  int b = gid / N_, n = gid % N_;
#pragma unroll
  for (int k = 0; k < 3; ++k)
    coorW[gid * 3 + k] = pcoor[((size_t)b * 3 + k) * N_ + n];
#pragma unroll 4
  for (int c = 0; c < C_; ++c)
    feaW[(size_t)gid * C_ + c] = pfea[((size_t)b * C_ + c) * N_ + n];
}

// ---------------------------------------------------------------------------
// 2) Farthest point sampling: one workgroup per batch; point cloud + running
//    distance kept entirely in LDS (128 KB of the 320 KB/WGP).
// ---------------------------------------------------------------------------
__global__ __launch_bounds__(1024) void fps_kernel(
    const float* __restrict__ coor, int* __restrict__ fps_idx) {
  __shared__ float sx[N_], sy[N_], sz[N_], sdist[N_];
  __shared__ unsigned long long red[32];
  __shared__ int sfar;
  const int b = blockIdx.x, t = threadIdx.x;
  const int lane = t & 31, wv = t >> 5;

  for (int i = t; i < N_; i += 1024) {
    const float* p = coor + ((size_t)b * N_ + i) * 3;
    sx[i] = p[0]; sy[i] = p[1]; sz[i] = p[2];
    sdist[i] = 1e10f;
  }
  __syncthreads();

  int far = 0;
  for (int j = 0; j < S_; ++j) {
    if (t == 0) fps_idx[b * S_ + j] = far;   // emit BEFORE update (ref semantics)
    const float cx = sx[far], cy = sy[far], cz = sz[far];
    unsigned long long key = 0ull;
#pragma unroll
    for (int u = 0; u < N_ / 1024; ++u) {
      int i = u * 1024 + t;
      float dx = sx[i] - cx, dy = sy[i] - cy, dz = sz[i] - cz;
      float d = sqrtf(dx * dx + dy * dy + dz * dz);
      d = fminf(sdist[i], d);
      sdist[i] = d;
      // max distance wins; tie -> smaller index (jnp.argmax first-occurrence)
      unsigned long long k2 =
          ((unsigned long long)__float_as_uint(d) << 32) |
          (unsigned long long)(0xFFFFFFFFu - (unsigned)i);
      if (k2 > key) key = k2;
    }
#pragma unroll
    for (int off = 16; off > 0; off >>= 1) {
      unsigned long long o = __shfl_xor(key, off, 32);
      if (o > key) key = o;
    }
    if (lane == 0) red[wv] = key;
    __syncthreads();
    if (wv == 0) {
      unsigned long long k2 = red[lane];
#pragma unroll
      for (int off = 16; off > 0; off >>= 1) {
        unsigned long long o = __shfl_xor(k2, off, 32);
        if (o > k2) k2 = o;
      }
      if (lane == 0)
        sfar = (int)(0xFFFFFFFFu - (unsigned)(k2 & 0xFFFFFFFFull));
    }
    __syncthreads();
    far = sfar;
  }
}

// ---------------------------------------------------------------------------
// 3) Gather sampled centroids; also write transposed coord output (B,3,S)
// ---------------------------------------------------------------------------
__global__ __launch_bounds__(256) void gather_centroids_kernel(
    const float* __restrict__ coor, const int* __restrict__ fps_idx,
    float* __restrict__ ncoor, float* __restrict__ outCoor) {
  int gid = blockIdx.x * blockDim.x + threadIdx.x;   // over B*S
  if (gid >= B_ * S_) return;
  int b = gid / S_, s = gid % S_;
  int idx = fps_idx[gid];
#pragma unroll
  for (int k = 0; k < 3; ++k) {
    float v = coor[((size_t)b * N_ + idx) * 3 + k];
    ncoor[gid * 3 + k] = v;
    outCoor[((size_t)b * 3 + k) * S_ + s] = v;
  }
}

// ---------------------------------------------------------------------------
// 4) Radius-masked 32-NN. Order within the group is irrelevant (max-pooled
//    downstream), so we only need the correct SET of 32 nearest + the nearest
//    ("first") substituted for out-of-radius slots. Per-thread candidate list
//    lives in LDS with [slot][tid] layout -> bank-conflict free rescans.
// ---------------------------------------------------------------------------
__global__ __launch_bounds__(128) void knn_kernel(
    const float* __restrict__ qCoor, const float* __restrict__ pCoor,
    int Np, float r2, int* __restrict__ gidx) {
  __shared__ float cd[G_][128];
  __shared__ int   ci[G_][128];
  const int t = threadIdx.x;
  const int q = blockIdx.x * 128 + t;       // B*S queries, S%128==0 -> b uniform
  const int b = q / S_;
  const float qx = qCoor[q * 3 + 0], qy = qCoor[q * 3 + 1], qz = qCoor[q * 3 + 2];
#pragma unroll
  for (int j = 0; j < G_; ++j) { cd[j][t] = 3.4e38f; ci[j][t] = 0; }
  float wmax = 3.4e38f; int wpos = 0;
  const float* base = pCoor + (size_t)b * Np * 3;
  for (int n = 0; n < Np; ++n) {
    float dx = base[n * 3 + 0] - qx;
    float dy = base[n * 3 + 1] - qy;
    float dz = base[n * 3 + 2] - qz;
    float d = dx * dx + dy * dy + dz * dz;
    if (d < wmax) {
      cd[wpos][t] = d; ci[wpos][t] = n;
      float nm = -1.f; int npos = 0;
#pragma unroll
      for (int j = 0; j < G_; ++j) {
        float v = cd[j][t];
        if (v > nm) { nm = v; npos = j; }
      }
      wmax = nm; wpos = npos;
    }
  }
  float best = 3.4e38f; int bidx = 0;
#pragma unroll
  for (int j = 0; j < G_; ++j) {
    float v = cd[j][t];
    if (v < best) { best = v; bidx = ci[j][t]; }
  }
#pragma unroll
  for (int j = 0; j < G_; ++j)
    gidx[(size_t)q * G_ + j] = (cd[j][t] > r2) ? bidx : ci[j][t];
}

// ---------------------------------------------------------------------------
// 5) Fused group-gather + MLP + ReLU + group-max, WMMA f32 16x16x4.
//    One wave per query point: all 32 lanes gather one neighbor row each into
//    LDS; W is DMA'd into LDS by the Tensor Data Mover (overlapped with the
//    gather); per 16-column tile both 16-row halves are accumulated, then the
//    group-max comes straight off the D-tile layout (per-lane max over the 8
//    accumulator VGPRs + shfl_xor(16)), fused with ReLU.
// ---------------------------------------------------------------------------
template <int CF>
__global__ __launch_bounds__(128) void mlp_max_kernel(
    const float* __restrict__ ptsCoor, int Np,
    const float* __restrict__ qCoor,
    const float* __restrict__ ptsFea,     // (B,Np,CF)
    const int*   __restrict__ gidx,       // (B,S,G)
    const float* __restrict__ Wt,         // (128, CF+3) row-major
    float invR,
    float* __restrict__ outFea) {         // (B,S,128)
  constexpr int CIN = CF + 3, CINP = CF + 4;   // pad K to multiple of 4
  __shared__ float gmat[4][32][CINP];
  __shared__ float wlds[C2_ * CIN];
  const int w = threadIdx.x >> 5, lane = threadIdx.x & 31;
  const int q = blockIdx.x * 4 + w;
  const int b = q / S_;
  const int lrow = lane & 15, khalf = lane >> 4;

  // Kick off the weight DMA (global -> LDS) before the gather.
  if (w == 0) tdm_load_1d((unsigned)(uintptr_t)wlds, Wt, (unsigned)(C2_ * CIN));

  // Gather: one neighbor row per lane.
  {
    const float qc0 = qCoor[q * 3 + 0], qc1 = qCoor[q * 3 + 1],
                qc2 = qCoor[q * 3 + 2];
    int idx = gidx[(size_t)q * G_ + lane];
    const float4* frow = (const float4*)(ptsFea + ((size_t)b * Np + idx) * CF);
#pragma unroll
    for (int c = 0; c < CF / 4; ++c) {
      float4 v = frow[c];
      gmat[w][lane][4 * c + 0] = v.x;
      gmat[w][lane][4 * c + 1] = v.y;
      gmat[w][lane][4 * c + 2] = v.z;
      gmat[w][lane][4 * c + 3] = v.w;
    }
    const float* pc = ptsCoor + ((size_t)b * Np + idx) * 3;
    gmat[w][lane][CF + 0] = (pc[0] - qc0) * invR;
    gmat[w][lane][CF + 1] = (pc[1] - qc1) * invR;
    gmat[w][lane][CF + 2] = (pc[2] - qc2) * invR;
    gmat[w][lane][CF + 3] = 0.f;
  }

  if (w == 0) __builtin_amdgcn_s_wait_tensorcnt(0);
  __syncthreads();

#pragma unroll 1
  for (int nt = 0; nt < 8; ++nt) {
    v8f acc0 = {};                     // neighbor rows 0..15
    v8f acc1 = {};                     // neighbor rows 16..31
    const int col = nt * 16 + lrow;
#pragma unroll
    for (int k = 0; k < CINP; k += 4) {
      int kk = k + khalf * 2;
      v2f a0; a0.x = gmat[w][lrow][kk];      a0.y = gmat[w][lrow][kk + 1];
      v2f a1; a1.x = gmat[w][16 + lrow][kk]; a1.y = gmat[w][16 + lrow][kk + 1];
      v2f bv;
      bv.x = (kk     < CIN) ? wlds[col * CIN + kk]     : 0.f;
      bv.y = (kk + 1 < CIN) ? wlds[col * CIN + kk + 1] : 0.f;
      acc0 = wmma4(a0, bv, acc0);
      acc1 = wmma4(a1, bv, acc1);
    }
    float m = fmaxf(acc0[0], acc1[0]);
#pragma unroll
    for (int r = 1; r < 8; ++r) m = fmaxf(m, fmaxf(acc0[r], acc1[r]));
    m = fmaxf(m, __shfl_xor(m, 16, 32));   // combine row halves
    m = fmaxf(m, 0.f);                     // ReLU
    if (lane < 16) outFea[(size_t)q * C2_ + nt * 16 + lane] = m;
  }
}

// ---------------------------------------------------------------------------
// 6) Dense layer Y = act(X * W^T [+ Res]) via WMMA f32 16x16x4.
//    One wave per 16x16 output tile; X rows / W rows stream through L2.
// ---------------------------------------------------------------------------
template <bool RELU, bool RES>
__global__ __launch_bounds__(256) void dense_kernel(
    const float* __restrict__ X,    // (M,K)
    const float* __restrict__ Wt,   // (Nc,K)
    const float* __restrict__ Res,  // (M,Nc) or nullptr
    float* __restrict__ Y,          // (M,Nc)
    int M, int K, int Nc) {
  const int w = threadIdx.x >> 5, lane = threadIdx.x & 31;
  const int lrow = lane & 15, khalf = lane >> 4;
  const int t = blockIdx.x * 8 + w;           // grid sized exactly
  const int ntiles = Nc >> 4;
  const int rt = t / ntiles, ct = t % ntiles;

  const float* xrow = X + (size_t)(rt * 16 + lrow) * K;
  const float* wrow = Wt + (size_t)(ct * 16 + lrow) * K;

  v8f acc = {};
  for (int k = 0; k < K; k += 4) {
    int kk = k + khalf * 2;
    v2f a;  a.x = xrow[kk];  a.y = xrow[kk + 1];
    v2f bv; bv.x = wrow[kk]; bv.y = wrow[kk + 1];
    acc = wmma4(a, bv, acc);
  }
#pragma unroll
  for (int r = 0; r < 8; ++r) {
    int row = rt * 16 + r + (lane < 16 ? 0 : 8);
    int col = ct * 16 + lrow;
    float v = acc[r];
    if (RES)  v += Res[(size_t)row * Nc + col];
    if (RELU) v = fmaxf(v, 0.f);
    Y[(size_t)row * Nc + col] = v;
  }
}

// ---------------------------------------------------------------------------
// 7) Final feature transpose to (B, C2, S)
// ---------------------------------------------------------------------------
__global__ __launch_bounds__(256) void write_fea_kernel(
    const float* __restrict__ fea, float* __restrict__ outFea) {
  int gid = blockIdx.x * blockDim.x + threadIdx.x;   // over B*S*C2
  if (gid >= B_ * S_ * C2_) return;
  int c = gid % C2_;
  int s = (gid / C2_) % S_;
  int b = gid / (S_ * C2_);
  outFea[((size_t)b * C2_ + c) * S_ + s] = fea[gid];
}

// ---------------------------------------------------------------------------
// Host-side orchestration
// ---------------------------------------------------------------------------
extern "C" void kernel_launch(void* const* d_in, const int* in_sizes, int n_in,
                              void* d_out, int out_size, void* d_ws,
                              size_t ws_size, hipStream_t stream) {
  const float* pcoor = (const float*)d_in[0];   // (B,3,N)
  const float* pfea  = (const float*)d_in[1];   // (B,C,N)
  const float* w_sa  = (const float*)d_in[2];   // (128,67)
  const float* w_la1 = (const float*)d_in[3];   // (128,131)
  const float* w1_1  = (const float*)d_in[4];   // (512,128)
  const float* w2_1  = (const float*)d_in[5];   // (128,512)
  const float* w_la2 = (const float*)d_in[6];   // (128,131)
  const float* w1_2  = (const float*)d_in[7];   // (512,128)
  const float* w2_2  = (const float*)d_in[8];   // (128,512)

  char* p = (char*)d_ws;
  auto carve = [&](size_t bytes) {
    char* r = p;
    p += (bytes + 255) & ~(size_t)255;
    return r;
  };
  float* coorW = (float*)carve((size_t)B_ * N_ * 3 * 4);
  float* feaW  = (float*)carve((size_t)B_ * N_ * C_ * 4);
  int*   fpsI  = (int*)  carve((size_t)B_ * S_ * 4);
  float* ncoor = (float*)carve((size_t)B_ * S_ * 3 * 4);
  int*   gidx1 = (int*)  carve((size_t)B_ * S_ * G_ * 4);
  int*   gidx2 = (int*)  carve((size_t)B_ * S_ * G_ * 4);
  float* feaA  = (float*)carve((size_t)B_ * S_ * C2_ * 4);
  float* feaB  = (float*)carve((size_t)B_ * S_ * C2_ * 4);
  float* feaC  = (float*)carve((size_t)B_ * S_ * C2_ * 4);
  float* hbuf  = (float*)carve((size_t)B_ * S_ * C2_ * EXP_ * 4);

  float* outCoor = (float*)d_out;                  // (B,3,S)
  float* outFea  = outCoor + (size_t)B_ * 3 * S_;  // (B,C2,S)

  const int M = B_ * S_;                // 8192 rows for dense layers
  const float r0sq = 0.2f * 0.2f, r12sq = 0.4f * 0.4f;
  const float invR0 = 1.0f / 0.2f, invR12 = 1.0f / 0.4f;

  prep_kernel<<<(B_ * N_ + 255) / 256, 256, 0, stream>>>(pcoor, pfea, coorW, feaW);
  fps_kernel<<<B_, 1024, 0, stream>>>(coorW, fpsI);
  gather_centroids_kernel<<<(B_ * S_ + 255) / 256, 256, 0, stream>>>(
      coorW, fpsI, ncoor, outCoor);

  knn_kernel<<<B_ * S_ / 128, 128, 0, stream>>>(ncoor, coorW, N_, r0sq, gidx1);
  knn_kernel<<<B_ * S_ / 128, 128, 0, stream>>>(ncoor, ncoor, S_, r12sq, gidx2);

  // Stage 1: set abstraction MLP (Cin = 64+3)
  mlp_max_kernel<64><<<B_ * S_ / 4, 128, 0, stream>>>(
      coorW, N_, ncoor, feaW, gidx1, w_sa, invR0, feaA);

  // Stage 2: inverted-residual block 1
  mlp_max_kernel<128><<<B_ * S_ / 4, 128, 0, stream>>>(
      ncoor, S_, ncoor, feaA, gidx2, w_la1, invR12, feaB);
  dense_kernel<true, false><<<(M / 16) * (C2_ * EXP_ / 16) / 8, 256, 0, stream>>>(
      feaB, w1_1, nullptr, hbuf, M, C2_, C2_ * EXP_);
  dense_kernel<true, true><<<(M / 16) * (C2_ / 16) / 8, 256, 0, stream>>>(
      hbuf, w2_1, feaA, feaC, M, C2_ * EXP_, C2_);

  // Stage 3: inverted-residual block 2 (same coords+radius -> reuse gidx2)
  mlp_max_kernel<128><<<B_ * S_ / 4, 128, 0, stream>>>(
      ncoor, S_, ncoor, feaC, gidx2, w_la2, invR12, feaB);
  dense_kernel<true, false><<<(M / 16) * (C2_ * EXP_ / 16) / 8, 256, 0, stream>>>(
      feaB, w1_2, nullptr, hbuf, M, C2_, C2_ * EXP_);
  dense_kernel<true, true><<<(M / 16) * (C2_ / 16) / 8, 256, 0, stream>>>(
      hbuf, w2_2, feaC, feaA, M, C2_ * EXP_, C2_);

  write_fea_kernel<<<(B_ * S_ * C2_ + 255) / 256, 256, 0, stream>>>(feaA, outFea);
}